// BarrelShifterRight48_43860206027315
// MI455X (gfx1250) — compile-verified
//
#include <hip/hip_runtime.h>
#include <stdint.h>

#define DATA_BITS       48
#define SHIFT_BITS      6
#define ROWS_PER_BLOCK  256
#define BLOCK_THREADS   256

typedef unsigned int u32x4 __attribute__((ext_vector_type(4)));
typedef int          i32x4 __attribute__((ext_vector_type(4)));
typedef int          i32x8 __attribute__((ext_vector_type(8)));

#if __has_builtin(__builtin_amdgcn_tensor_load_to_lds)
#define HAVE_TDM 1
#else
#define HAVE_TDM 0
#endif

#if HAVE_TDM
// 1-D contiguous TDM copy: nelem float32 from global -> LDS byte offset.
// D# field layout per CDNA5 ISA ch.8 (group0: count/lds_addr/global_addr/type,
// group1: data_size/tensor_dim0/tile_dim0/strides). Groups 2/3 zero (<=2D).
__device__ __forceinline__ void tdm_copy_1d(uint32_t lds_byte, const float* gsrc,
                                            uint32_t nelem /* < 65536 */) {
    uint64_t ga = (uint64_t)(uintptr_t)gsrc;
    u32x4 g0;
    g0.x = 1u;                                                 // count=1 (valid), user mode
    g0.y = lds_byte;                                           // lds_addr
    g0.z = (uint32_t)ga;                                       // global_addr[31:0]
    g0.w = ((uint32_t)(ga >> 32) & 0x01FFFFFFu) | (2u << 30);  // global_addr[56:32], type=2
    i32x8 g1;
    g1[0] = (int)(2u << 16);                                   // data_size=2 -> 4 bytes
    g1[1] = (int)((nelem & 0xFFFFu) << 16);                    // tensor_dim0[15:0]
    g1[2] = (int)(((nelem >> 16) & 0xFFFFu) | (1u << 16));     // tensor_dim0[31:16], tensor_dim1=1
    g1[3] = (int)((nelem & 0xFFFFu) << 16);                    // tile_dim0 = nelem
    g1[4] = 1;                                                 // tile_dim1=1, tile_dim2=0 (unused)
    g1[5] = (int)nelem;                                        // tensor_dim0_stride[31:0]
    g1[6] = 0;
    g1[7] = 0;
    i32x4 gz = {0, 0, 0, 0};
#if defined(__clang_major__) && __clang_major__ >= 23
    i32x8 gz8 = {0, 0, 0, 0, 0, 0, 0, 0};
    __builtin_amdgcn_tensor_load_to_lds(g0, g1, gz, gz, gz8, 0);
#else
    __builtin_amdgcn_tensor_load_to_lds(g0, g1, gz, gz, 0);
#endif
}
#endif

__global__ __launch_bounds__(BLOCK_THREADS)
void BarrelShifterRight48_kernel(const float* __restrict__ X,
                                 const float* __restrict__ S,
                                 float* __restrict__ out,
                                 int nrows) {
    __shared__ float lds_x[ROWS_PER_BLOCK * DATA_BITS];   // 48 KB
    __shared__ float lds_s[ROWS_PER_BLOCK * SHIFT_BITS];  //  6 KB

    const uint32_t tid  = threadIdx.x;
    const uint64_t row0 = (uint64_t)blockIdx.x * ROWS_PER_BLOCK;

#if HAVE_TDM
    if (tid < 32) {  // wave-uniform branch: only wave 0 issues the DMAs
        tdm_copy_1d((uint32_t)(uintptr_t)&lds_x[0],
                    X + row0 * DATA_BITS, ROWS_PER_BLOCK * DATA_BITS);
        tdm_copy_1d((uint32_t)(uintptr_t)&lds_s[0],
                    S + row0 * SHIFT_BITS, ROWS_PER_BLOCK * SHIFT_BITS);
        __builtin_amdgcn_s_wait_tensorcnt(0);
    }
#else
    for (uint32_t i = tid; i < ROWS_PER_BLOCK * DATA_BITS; i += BLOCK_THREADS)
        lds_x[i] = X[row0 * DATA_BITS + i];
    for (uint32_t i = tid; i < ROWS_PER_BLOCK * SHIFT_BITS; i += BLOCK_THREADS)
        lds_s[i] = S[row0 * SHIFT_BITS + i];
#endif
    __syncthreads();

    const uint32_t lane = tid & 31u;
    const uint32_t wave = tid >> 5;

    // Per-lane shift amount for row (wave*32 + lane). Stride-6 dword access:
    // banks 6*L mod 64 distinct for L=0..31 -> conflict-free.
    uint32_t n = 0;
    {
        const float* sp = &lds_s[(wave * 32 + lane) * SHIFT_BITS];
#pragma unroll
        for (int k = 0; k < SHIFT_BITS; ++k)
            n += (sp[k] > 0.5f) ? (32u >> k) : 0u;
    }

    float my_sticky = 0.0f;
    const uint64_t sticky_base = (uint64_t)nrows * DATA_BITS;
    const uint64_t out_row0 = (row0 + wave * 32u) * DATA_BITS;

#pragma unroll 4
    for (int r = 0; r < 32; ++r) {
        const int row_l = (int)(wave * 32u) + r;
        const float* xp = &lds_x[row_l * DATA_BITS];

        // Lane L reads column L (consecutive LDS addresses -> no bank conflicts).
        float a = xp[lane];                 // columns 0..31
        float b = xp[32 + (lane & 15)];     // columns 32..47 (lanes 16..31 re-read: broadcast)

        // ballot bit L = column L; MSB-first -> bit-reverse into V.
#if __has_builtin(__builtin_amdgcn_ballot_w32)
        uint32_t m_hi = __builtin_amdgcn_ballot_w32(a > 0.5f);
        uint32_t m_lo = __builtin_amdgcn_ballot_w32((lane < 16) && (b > 0.5f));
#else
        uint32_t m_hi = (uint32_t)__ballot(a > 0.5f);
        uint32_t m_lo = (uint32_t)__ballot((lane < 16) && (b > 0.5f));
#endif
        uint64_t V = ((uint64_t)__brev(m_hi) << 16) | (uint64_t)(__brev(m_lo) >> 16);

#if __has_builtin(__builtin_amdgcn_readlane)
        uint32_t nr = (uint32_t)__builtin_amdgcn_readlane((int)n, r);
#else
        uint32_t nr = (uint32_t)__shfl((int)n, r, 32);
#endif
        uint64_t W       = V >> nr;                      // nr in [0,63]; V < 2^48
        uint64_t dropped = V & ((1ull << nr) - 1ull);    // OR of shifted-out bits
        float st = dropped ? 1.0f : 0.0f;
        my_sticky = (lane == (uint32_t)r) ? st : my_sticky;

        // Unpack: output col c = bit (47-c) of W.  hi = W[47:16], lo16 = W[15:0].
        uint32_t hi   = (uint32_t)(W >> 16);
        uint32_t lo16 = (uint32_t)W & 0xFFFFu;

        float* op = out + out_row0 + (uint64_t)r * DATA_BITS;
        __builtin_nontemporal_store((float)((hi >> (31 - lane)) & 1u), op + lane);
        if (lane < 16)
            __builtin_nontemporal_store((float)((lo16 >> (15 - lane)) & 1u), op + 32 + lane);
    }

    // Lane L holds sticky of row (wave*32 + L): one fully coalesced store.
    __builtin_nontemporal_store(my_sticky,
        out + sticky_base + row0 + (uint64_t)tid);
}

extern "C" void kernel_launch(void* const* d_in, const int* in_sizes, int n_in,
                              void* d_out, int out_size, void* d_ws, size_t ws_size,
                              hipStream_t stream) {
    const float* X = (const float*)d_in[0];   // [BATCH, 48] float32
    const float* S = (const float*)d_in[1];   // [BATCH, 6]  float32
    float* out = (float*)d_out;               // [BATCH*48] shifted ++ [BATCH] sticky

    int nrows  = in_sizes[0] / DATA_BITS;     // 1048576
    int blocks = nrows / ROWS_PER_BLOCK;      // 4096 (exact for BATCH=2^20)
    BarrelShifterRight48_kernel<<<blocks, BLOCK_THREADS, 0, stream>>>(X, S, out, nrows);
}